// GATLayer_85220741087239
// MI455X (gfx1250) — compile-verified
//
#include <hip/hip_runtime.h>
#include <cstddef>

typedef float v2f __attribute__((ext_vector_type(2)));
typedef float v4f __attribute__((ext_vector_type(4)));
typedef float v8f __attribute__((ext_vector_type(8)));
typedef int   v4i __attribute__((ext_vector_type(4)));

#define B_  8
#define N_  2048
#define C_  128
#define D_  128
#define NEG_BIG (-9.0e15f)

static __device__ __forceinline__ v8f wmma_f32_4(v2f a, v2f b, v8f c) {
    // V_WMMA_F32_16X16X4_F32 : D = A(16x4,f32) * B(4x16,f32) + C(16x16,f32)
    return __builtin_amdgcn_wmma_f32_16x16x4_f32(false, a, false, b, (short)0, c,
                                                 false, false);
}

// ---------------------------------------------------------------------------
// Kernel 1: WhT[b][d][j] = sum_c W[d][c] * x[b][j][c]   (D = W * x^T)
// grid = B * (N/16) blocks, 256 threads = 8 waves; wave w handles d-tile w.
// WMMA roles: A = W tile (M=d), B = x tile (N=j).
// ---------------------------------------------------------------------------
__global__ __launch_bounds__(256)
void wht_kernel(const float* __restrict__ x, const float* __restrict__ W,
                float* __restrict__ wht) {
    const int bid  = blockIdx.x;
    const int b    = bid >> 7;          // / (N/16)
    const int jt   = bid & 127;
    const int tid  = threadIdx.x;
    const int wv   = tid >> 5;          // d-tile
    const int lane = tid & 31;
    const int nl   = lane & 15;
    const int kh   = (lane >> 4) << 1;  // K offset 0 or 2

    const float* xrow = x + ((size_t)(b * N_ + jt * 16 + nl)) * C_;
    const float* wrow = W + (size_t)(wv * 16 + nl) * C_;

    v8f acc = {};
#pragma unroll 4
    for (int k0 = 0; k0 < C_; k0 += 4) {
        const int kb = k0 + kh;
        v2f a = *(const v2f*)(wrow + kb);   // A: lane m = d row, K = kb, kb+1
        v2f bm = *(const v2f*)(xrow + kb);  // B: lane n = j col, K = kb, kb+1
        acc = wmma_f32_4(a, bm, acc);
    }

    // C/D layout: VGPR r -> (M = r | r+8, N = lane&15). Store WhT row-major.
    float* ob = wht + (size_t)b * D_ * N_ + (size_t)jt * 16 + nl;
#pragma unroll
    for (int r = 0; r < 8; ++r) {
        const int d = wv * 16 + r + ((lane >> 4) << 3);
        ob[(size_t)d * N_] = acc[r];
        // advance handled by d in index
    }
}

// ---------------------------------------------------------------------------
// Kernel 2: f1[b][j] = sum_d WhT[b][d][j]*a[d];  f2 likewise with a[D..2D)
// ---------------------------------------------------------------------------
__global__ __launch_bounds__(256)
void f_kernel(const float* __restrict__ wht, const float* __restrict__ a,
              float* __restrict__ f1, float* __restrict__ f2) {
    const int t = blockIdx.x * 256 + threadIdx.x;   // 0 .. B*N-1
    const int b = t >> 11;
    const int j = t & (N_ - 1);
    const float* wb = wht + (size_t)b * D_ * N_ + j;
    float s1 = 0.f, s2 = 0.f;
#pragma unroll 4
    for (int d = 0; d < D_; ++d) {
        const float w = wb[(size_t)d * N_];
        s1 += w * a[d];
        s2 += w * a[D_ + d];
    }
    f1[t] = s1;
    f2[t] = s2;
}

// ---------------------------------------------------------------------------
// Kernel 3: fused masked-softmax attention + (attn @ Wh) + ELU.
// One block (8 waves) per (batch, 16-row i-tile).
// Pass 1: pack adj bits to LDS, masked max of f2 per row (exact row max via
//         monotonicity of leaky_relu).  Pass 2: waves split j; p-fragments go
//         straight into V_WMMA_F32_16X16X4_F32 against WhT B-fragments.
// ---------------------------------------------------------------------------
__global__ __launch_bounds__(256)
void gat_kernel(const float* __restrict__ wht, const float* __restrict__ f1g,
                const float* __restrict__ f2g, const int* __restrict__ adj,
                float* __restrict__ out) {
    __shared__ unsigned lds_bits[16 * 65];       // adj bit-mask, padded stride
    __shared__ float lds_mw[256];                // per-thread max partials
    __shared__ float lds_m[16], lds_f1[16], lds_pone[16], lds_l[16];
    __shared__ float lds_lw[8 * 16];             // per-wave row sums
    __shared__ float accbuf[4 * 8 * 32 * 8];     // 32KB tree-reduce buffer

    const int bid  = blockIdx.x;
    const int b    = bid >> 7;
    const int i0   = (bid & 127) * 16;
    const int tid  = threadIdx.x;
    const int wv   = tid >> 5;
    const int lane = tid & 31;
    const int nl   = lane & 15;

    const float* f2b  = f2g + b * N_;
    const float* whtb = wht + (size_t)b * D_ * N_;

    // ---------------- Pass 1: bits + masked row max --------------------------
    {
        const int i   = tid >> 4;         // row 0..15
        const int seg = tid & 15;         // 128-wide j segment
        const int*   arow = adj + (size_t)(i0 + i) * N_ + seg * 128;
        const float* frow = f2b + seg * 128;
        float mmax = -3.0e38f;
#pragma unroll
        for (int w2 = 0; w2 < 4; ++w2) {
            unsigned bits = 0u;
#pragma unroll
            for (int q = 0; q < 8; ++q) {
                v4i av = *(const v4i*)(arow + w2 * 32 + q * 4);
                v4f fv = *(const v4f*)(frow + w2 * 32 + q * 4);
                if (av[0]) { bits |= 1u << (q * 4 + 0); mmax = fmaxf(mmax, fv[0]); }
                if (av[1]) { bits |= 1u << (q * 4 + 1); mmax = fmaxf(mmax, fv[1]); }
                if (av[2]) { bits |= 1u << (q * 4 + 2); mmax = fmaxf(mmax, fv[2]); }
                if (av[3]) { bits |= 1u << (q * 4 + 3); mmax = fmaxf(mmax, fv[3]); }
            }
            lds_bits[i * 65 + seg * 4 + w2] = bits;
        }
        lds_mw[i * 16 + seg] = mmax;
    }
    __syncthreads();
    if (tid < 16) {
        float M = -3.0e38f;
#pragma unroll
        for (int s = 0; s < 16; ++s) M = fmaxf(M, lds_mw[tid * 16 + s]);
        const bool empty = (M < -1.0e38f);
        const float f1v  = f1g[b * N_ + i0 + tid];
        const float sum  = f1v + M;
        const float emax = fmaxf(sum, 0.2f * sum);   // leaky_relu, slope 0.2
        lds_m[tid]    = empty ? 0.0f : emax;
        lds_pone[tid] = empty ? 1.0f : 0.0f;         // uniform-softmax fallback
        lds_f1[tid]   = f1v;
    }
    __syncthreads();

    // ---------------- Pass 2: p-fragments -> WMMA accumulate ----------------
    const float f1i = lds_f1[nl];
    const float mi  = lds_m[nl];
    const float pm  = lds_pone[nl];
    const float* wbase = whtb + (size_t)nl * N_;   // d = dt*16 + nl row base

    v8f acc[8] = {};
    float lsum = 0.0f;

    for (int g = 0; g < 64; ++g) {                 // this wave's 64 j-groups
        const int j0 = g * 32 + wv * 4;
        const int jA = j0 + ((lane >> 4) << 1);    // K-half offset
        const unsigned word = lds_bits[nl * 65 + (jA >> 5)];
        const unsigned sh   = (unsigned)(jA & 31);

        const v2f f2v = *(const v2f*)(f2b + jA);
        float s0 = f1i + f2v[0]; s0 = fmaxf(s0, 0.2f * s0);
        float s1 = f1i + f2v[1]; s1 = fmaxf(s1, 0.2f * s1);
        const float p0 = ((word >> sh) & 1u)        ? __expf(s0 - mi) : pm;
        const float p1 = ((word >> (sh + 1)) & 1u)  ? __expf(s1 - mi) : pm;

        v2f A; A[0] = p0; A[1] = p1;               // A-fragment (rows=i, K=j)
        lsum += p0 + p1;

#pragma unroll
        for (int dt = 0; dt < 8; ++dt) {
            const v2f Bv = *(const v2f*)(wbase + (size_t)dt * 16 * N_ + jA);
            acc[dt] = wmma_f32_4(A, Bv, acc[dt]);
        }
    }

    // ---------------- row-sum reduction (deterministic) ---------------------
    float lpair = lsum + __shfl_xor(lsum, 16);
    if (lane < 16) lds_lw[wv * 16 + lane] = lpair;
    __syncthreads();
    if (tid < 16) {
        float t = 0.0f;
#pragma unroll
        for (int w2 = 0; w2 < 8; ++w2) t += lds_lw[w2 * 16 + tid];
        lds_l[tid] = t;
    }
    __syncthreads();

    // ---------------- accumulator tree reduction across waves ---------------
    for (int half = 4; half >= 1; half >>= 1) {
        if (wv >= half && wv < 2 * half) {
            const int slot = wv - half;
#pragma unroll
            for (int dt = 0; dt < 8; ++dt)
                *(v8f*)&accbuf[(((size_t)slot * 8 + dt) * 32 + lane) * 8] = acc[dt];
        }
        __syncthreads();
        if (wv < half) {
#pragma unroll
            for (int dt = 0; dt < 8; ++dt)
                acc[dt] = acc[dt] +
                    *(const v8f*)&accbuf[(((size_t)wv * 8 + dt) * 32 + lane) * 8];
        }
        __syncthreads();
    }

    // ---------------- wave 0: normalize, ELU, coalesced store ---------------
    if (wv == 0) {
        float* ob = out + ((size_t)b * N_ + i0) * D_;
#pragma unroll
        for (int dt = 0; dt < 8; ++dt) {
#pragma unroll
            for (int r = 0; r < 8; ++r) {
                const int row = r + ((lane >> 4) << 3);
                const float li = lds_l[row];
                const float h  = acc[dt][r] / li;
                const float o  = (h > 0.0f) ? h : (__expf(h) - 1.0f);
                ob[(size_t)row * D_ + dt * 16 + nl] = o;
            }
        }
    }
}

// ---------------------------------------------------------------------------
// Launch: ws layout = [ WhT : B*D*N | f1 : B*N | f2 : B*N ] floats (~8.2 MB)
// ---------------------------------------------------------------------------
extern "C" void kernel_launch(void* const* d_in, const int* in_sizes, int n_in,
                              void* d_out, int out_size, void* d_ws, size_t ws_size,
                              hipStream_t stream) {
    const float* x   = (const float*)d_in[0];
    const int*   adj = (const int*)d_in[1];
    const float* W   = (const float*)d_in[2];
    const float* a   = (const float*)d_in[3];
    float* out = (float*)d_out;

    float* wht = (float*)d_ws;                       // B*D*N   = 2,097,152
    float* f1  = wht + (size_t)B_ * D_ * N_;         // B*N     = 16,384
    float* f2  = f1 + (size_t)B_ * N_;               // B*N     = 16,384

    wht_kernel<<<dim3(B_ * (N_ / 16)), dim3(256), 0, stream>>>(x, W, wht);
    f_kernel<<<dim3((B_ * N_) / 256), dim3(256), 0, stream>>>(wht, a, f1, f2);
    gat_kernel<<<dim3(B_ * (N_ / 16)), dim3(256), 0, stream>>>(wht, f1, f2, adj, out);
}